// LapLoss_77807627534603
// MI455X (gfx1250) — compile-verified
//
#include <hip/hip_runtime.h>

// Laplacian pyramid L1 loss, MI455X (gfx1250).
// Linearity: lap(pred)-lap(target) == lap(pred-target)  -> one pyramid.
// 5-tap Gaussian convs expressed as banded matmuls on V_WMMA_F32_16X16X4_F32.
// Round 2: branchless per-lane weight constants (pure v_cndmask, no exec
// branching between WMMAs) + register-prefetched A/B fragments so the 9-deep
// WMMA accumulation chain issues back-to-back.

typedef __attribute__((__ext_vector_type__(2))) float v2f;
typedef __attribute__((__ext_vector_type__(8))) float v8f;

__device__ __forceinline__ int imin(int a, int b) { return a < b ? a : b; }

// Gaussian taps [1,4,6,4,1]/16 — branchless select chain (no ||/&&).
__device__ __forceinline__ float w5(int t) {
  float r = 0.0f;
  r = (t == 0) ? 0.0625f : r;
  r = (t == 1) ? 0.25f   : r;
  r = (t == 2) ? 0.375f  : r;
  r = (t == 3) ? 0.25f   : r;
  r = (t == 4) ? 0.0625f : r;
  return r;
}

// Upsample (zero-stuff + 4*kernel) banded weights, tile-local:
// even fine y: coarse m = y/2 + {0,1,2} w {1/8, 3/4, 1/8}
// odd  fine y: coarse m = (y+1)/2 + {0,1} w {1/2, 1/2}
// Branchless: bitwise bool combines only.
__device__ __forceinline__ float uw(int y, int m) {
  const int odd = y & 1;
  const int dE = m - (y >> 1);
  const int dO = m - ((y + 1) >> 1);
  float r = 0.0f;
  r = ((!odd) & (dE == 0)) ? 0.125f : r;
  r = ((!odd) & (dE == 1)) ? 0.75f  : r;
  r = ((!odd) & (dE == 2)) ? 0.125f : r;
  r = (odd & (dO == 0))    ? 0.5f   : r;
  r = (odd & (dO == 1))    ? 0.5f   : r;
  return r;
}

// jnp.pad 'reflect' index mapping (single reflection + clamp for deep OOB
// that only feeds store-guarded tiles).
__device__ __forceinline__ int refl(int i, int n) {
  if (i < 0) i = -i;
  if (i >= n) i = 2 * n - 2 - i;
  if (i < 0) i = 0;
  if (i >= n) i = n - 1;
  return i;
}

__device__ __forceinline__ v8f wmma16x16x4(v2f a, v2f b, v8f c) {
  return __builtin_amdgcn_wmma_f32_16x16x4_f32(false, a, false, b, (short)0, c,
                                               false, false);
}

// ---------------- blur + decimate: dn = gauss(d)[::2, ::2] -----------------
// Block: 256 threads (8 waves), covers a 32x64 down-tile (8 x 16x16 tiles).
#define BD_ROWS 32
#define BD_COLS 64
#define IN_R 67    // 2*32+3 halo rows (fine space)
#define IN_C 131   // 2*64+3 halo cols
#define IN_CP 132
#define TV_CP 136  // tmpV padded stride (A-chunks read up to col 131)

__global__ __launch_bounds__(256) void k_blur_down(
    const float* __restrict__ inA, const float* __restrict__ inB,
    float* __restrict__ out, int S, int Sd) {
  __shared__ float s_in[IN_R * IN_CP];
  __shared__ float s_tmp[BD_ROWS * TV_CP];
  const int tid = threadIdx.x;
  const int img = blockIdx.z;
  const int R0 = blockIdx.y * BD_ROWS;  // down space
  const int C0 = blockIdx.x * BD_COLS;
  const long ibase = (long)img * S * S;

  const int lane = tid & 31;
  const int half = lane >> 4, n = lane & 15;

  // Per-lane B-matrix constants H2[j,x] = w5(j - 2x), chunk c rows 4c+2*half.
  // Pure cndmask chains; computed up front to overlap with staging.
  v2f bH[9];
#pragma unroll
  for (int c = 0; c < 9; ++c) {
    const int k0 = 4 * c + 2 * half;
    bH[c].x = w5(k0 - 2 * n);
    bH[c].y = w5(k0 + 1 - 2 * n);
  }

  // Stage fine-space halo (reflect padded); L0 computes pred-target inline.
  for (int idx = tid; idx < IN_R * IN_C; idx += 256) {
    int r = idx / IN_C, c = idx - r * IN_C;
    long g = ibase + (long)refl(2 * R0 - 2 + r, S) * S + refl(2 * C0 - 2 + c, S);
    float v = inA[g];
    if (inB) v -= inB[g];
    s_in[r * IN_CP + c] = v;
  }
  __syncthreads();

  // Vertical 5-tap at even rows only (VALU): tmp[i][j] = sum_t w[t]*in[2i+t][j]
  for (int idx = tid; idx < BD_ROWS * IN_C; idx += 256) {
    int i = idx / IN_C, j = idx - i * IN_C;
    const float* p = &s_in[(2 * i) * IN_CP + j];
    s_tmp[i * TV_CP + j] = 0.0625f * (p[0] + p[4 * IN_CP]) +
                           0.25f * (p[IN_CP] + p[3 * IN_CP]) +
                           0.375f * p[2 * IN_CP];
  }
  __syncthreads();

  // Horizontal decimating conv as banded matmul:
  // down(16x16) = tmpV(16x36) x H2(36x16). Prefetch A frags, then 9 WMMAs.
  const int wv = tid >> 5;
  const int tr = wv >> 2, tc = wv & 3;
  const float* tbase = &s_tmp[(tr * 16 + n) * TV_CP + tc * 32];
  v2f aF[9];
#pragma unroll
  for (int c = 0; c < 9; ++c) {
    const int k0 = 4 * c + 2 * half;
    aF[c].x = tbase[k0];
    aF[c].y = tbase[k0 + 1];
  }
  v8f acc = {0.f, 0.f, 0.f, 0.f, 0.f, 0.f, 0.f, 0.f};
#pragma unroll
  for (int c = 0; c < 9; ++c) acc = wmma16x16x4(aF[c], bH[c], acc);

#pragma unroll
  for (int k = 0; k < 8; ++k) {
    int rd = R0 + tr * 16 + k + half * 8;  // D: VGPR k -> row k | k+8
    int cd = C0 + tc * 16 + n;
    if (rd < Sd && cd < Sd) out[(long)img * Sd * Sd + (long)rd * Sd + cd] = acc[k];
  }
}

// --------- laplacian residual: sum |d - upsample(dn)| per block ------------
// Block: 256 threads, 32x64 fine tile. up = U(16x12) x Dn(12x16) x V(16x16).
#define CR 18
#define CC 34
#define CCP 36

__global__ __launch_bounds__(256) void k_lap_abs(
    const float* __restrict__ inA, const float* __restrict__ inB,
    const float* __restrict__ dn, float* __restrict__ partial, int S, int Sc) {
  __shared__ float s_c[CR * CCP];
  __shared__ float s_p[8][16 * 17];
  __shared__ float s_red[256];
  const int tid = threadIdx.x;
  const int img = blockIdx.z;
  const int R0 = blockIdx.y * 32;
  const int C0 = blockIdx.x * 64;
  const int I0 = R0 >> 1, J0 = C0 >> 1;

  const int lane = tid & 31;
  const int half = lane >> 4, n = lane & 15;

  // uC[c] = {uw(n, 4c+2h), uw(n, 4c+2h+1)}; serves as A in pass A (c<3) and
  // as B (= V[k,n] = uw(n,k)) in pass B (c<4). Branchless, hoisted.
  v2f uC[4];
#pragma unroll
  for (int c = 0; c < 4; ++c) {
    const int k0 = 4 * c + 2 * half;
    uC[c].x = uw(n, k0);
    uC[c].y = uw(n, k0 + 1);
  }

  // Stage coarse halo rows I0-1..I0+16, cols J0-1..J0+32 (reflect).
  for (int idx = tid; idx < CR * CC; idx += 256) {
    int r = idx / CC, c = idx - r * CC;
    s_c[r * CCP + c] =
        dn[(long)img * Sc * Sc + (long)refl(I0 - 1 + r, Sc) * Sc + refl(J0 - 1 + c, Sc)];
  }
  __syncthreads();

  const int wv = tid >> 5;
  const int tr = wv >> 2, tc = wv & 3;

  // Pass A: P = U x Dn (3 WMMAs). Columns n>=10 are garbage; V zeroes them.
  const int cc = imin(8 * tc + n, CC - 1);
  v2f bD[3];
#pragma unroll
  for (int c = 0; c < 3; ++c) {
    const int k0 = 4 * c + 2 * half;
    bD[c].x = s_c[imin(8 * tr + k0, CR - 1) * CCP + cc];      // B[K=k0, N=n]
    bD[c].y = s_c[imin(8 * tr + k0 + 1, CR - 1) * CCP + cc];
  }
  v8f P = {0.f, 0.f, 0.f, 0.f, 0.f, 0.f, 0.f, 0.f};
#pragma unroll
  for (int c = 0; c < 3; ++c) P = wmma16x16x4(uC[c], bD[c], P);

  // In-wave LDS transpose of P (D layout -> A layout).
  float* pt = &s_p[wv][0];
#pragma unroll
  for (int k = 0; k < 8; ++k) pt[(k + half * 8) * 17 + n] = P[k];

  // Pass B: up = P x V (4 WMMAs). V[k,n] = uw(n,k) = uC.
  v2f aP[4];
#pragma unroll
  for (int c = 0; c < 4; ++c) {
    const int k0 = 4 * c + 2 * half;
    aP[c].x = pt[n * 17 + k0];
    aP[c].y = pt[n * 17 + k0 + 1];
  }
  v8f up = {0.f, 0.f, 0.f, 0.f, 0.f, 0.f, 0.f, 0.f};
#pragma unroll
  for (int c = 0; c < 4; ++c) up = wmma16x16x4(aP[c], uC[c], up);

  // |d - up|, accumulate.
  float local = 0.0f;
#pragma unroll
  for (int k = 0; k < 8; ++k) {
    int rf = R0 + tr * 16 + k + half * 8;
    int cf = C0 + tc * 16 + n;
    if (rf < S && cf < S) {
      long g = (long)img * S * S + (long)rf * S + cf;
      float dv = inA[g];
      if (inB) dv -= inB[g];
      local += fabsf(dv - up[k]);
    }
  }
  s_red[tid] = local;
  __syncthreads();
  for (int s = 128; s > 0; s >>= 1) {
    if (tid < s) s_red[tid] += s_red[tid + s];
    __syncthreads();
  }
  if (tid == 0) {
    int bl = (blockIdx.z * gridDim.y + blockIdx.y) * gridDim.x + blockIdx.x;
    partial[bl] = s_red[0];
  }
}

// ------------------- deterministic final reduction -------------------------
__global__ __launch_bounds__(256) void k_finalize(const float* __restrict__ partial,
                                                  float* __restrict__ out) {
  __shared__ float s_red[256];
  const int cnt[5] = {6144, 1536, 384, 96, 48};
  const int off[5] = {0, 6144, 7680, 8064, 8160};
  const float scale[5] = {1.0f / 12582912.0f, 1.0f / 3145728.0f, 1.0f / 786432.0f,
                          1.0f / 196608.0f, 1.0f / 49152.0f};
  const int tid = threadIdx.x;
  float loss = 0.0f;
  for (int l = 0; l < 5; ++l) {
    float s = 0.0f;
    for (int i = tid; i < cnt[l]; i += 256) s += partial[off[l] + i];
    s_red[tid] = s;
    __syncthreads();
    for (int st = 128; st > 0; st >>= 1) {
      if (tid < st) s_red[tid] += s_red[tid + st];
      __syncthreads();
    }
    if (tid == 0) loss += s_red[0] * scale[l];
    __syncthreads();
  }
  if (tid == 0) out[0] = loss;
}

extern "C" void kernel_launch(void* const* d_in, const int* in_sizes, int n_in,
                              void* d_out, int out_size, void* d_ws, size_t ws_size,
                              hipStream_t stream) {
  (void)in_sizes; (void)n_in; (void)out_size; (void)ws_size;
  const float* pred = (const float*)d_in[0];
  const float* targ = (const float*)d_in[1];
  float* ws = (float*)d_ws;
  const long nImg = 48;  // B*C = 16*3

  // ws float layout: d1(256^2) d2(128^2) d3(64^2) d4(32^2) d5(16^2) partials
  size_t off[6];
  off[1] = 0;
  off[2] = off[1] + (size_t)nImg * 256 * 256;
  off[3] = off[2] + (size_t)nImg * 128 * 128;
  off[4] = off[3] + (size_t)nImg * 64 * 64;
  off[5] = off[4] + (size_t)nImg * 32 * 32;
  float* partial = ws + off[5] + (size_t)nImg * 16 * 16;
  const int pOff[5] = {0, 6144, 7680, 8064, 8160};

  for (int L = 0; L < 5; ++L) {
    const int S = 512 >> L, Sd = S >> 1;
    const float* inA = L ? ws + off[L] : pred;
    const float* inB = L ? nullptr : targ;
    float* dnp = ws + off[L + 1];
    dim3 g1((Sd + 63) / 64, (Sd + 31) / 32, 48);
    k_blur_down<<<g1, 256, 0, stream>>>(inA, inB, dnp, S, Sd);
    dim3 g2((S + 63) / 64, (S + 31) / 32, 48);
    k_lap_abs<<<g2, 256, 0, stream>>>(inA, inB, dnp, partial + pOff[L], S, Sd);
  }
  k_finalize<<<1, 256, 0, stream>>>(partial, (float*)d_out);
}